// gnn_block_22737556865354
// MI455X (gfx1250) — compile-verified
//
#include <hip/hip_runtime.h>
#include <stdint.h>

// NOTE: grid sizes below rely on these exact values dividing evenly:
//   N_EDGES % 128 == 0, % 16 == 0, (N_EDGES*8) % 256 == 0,
//   (N_NODES*8) % 256 == 0, N_NODES % 8 == 0.
#define N_NODES 100000
#define N_EDGES 1600000
#define NEG_SLOPE 0.2f
#define LN_EPS 1e-5f

typedef float v2f __attribute__((ext_vector_type(2)));
typedef float v8f __attribute__((ext_vector_type(8)));

// Only enforce on the device pass: the host pass can call aux-target builtins
// inside __device__ code but __has_builtin() reports 0 for them.
#if defined(__HIP_DEVICE_COMPILE__) && \
    !__has_builtin(__builtin_amdgcn_wmma_f32_16x16x4_f32)
#error "missing __builtin_amdgcn_wmma_f32_16x16x4_f32 on this toolchain"
#endif

__device__ __forceinline__ v8f wmma_k4(v2f a, v2f b, v8f c) {
  // D = A(16x4,f32) * B(4x16,f32) + C(16x16,f32)
  return __builtin_amdgcn_wmma_f32_16x16x4_f32(false, a, false, b, (short)0, c,
                                               false, false);
}

// ---------------------------------------------------------------------------
// GEMM: out[r,0:128] = epilogue(A[r,0:128] @ W[128,128])
// Block = 256 threads (8 waves). Block covers 64 rows; wave w owns col tile w.
// Dynamic LDS: sW[128][128] + sA[64][132]  (~97 KB)
// N_NODES is NOT a multiple of 64, so this kernel keeps its row guards.
// ---------------------------------------------------------------------------
#define EPI_NONE 0
#define EPI_GELU 1
#define EPI_RES 2

template <int EPI>
__global__ __launch_bounds__(256) void gemm128_kernel(
    const float* __restrict__ A, const float* __restrict__ W,
    const float* __restrict__ bias, const float* __restrict__ res,
    float* __restrict__ out, int nrows) {
  extern __shared__ float smem[];
  float* sW = smem;              // 128*128
  float* sA = smem + 128 * 128;  // 64*132 (pad 4 -> conflict-free A frags)
  const int tid = threadIdx.x;
  const long long row0 = (long long)blockIdx.x * 64;

  for (int i = tid; i < 128 * 128; i += 256) sW[i] = W[i];
  for (int i = tid; i < 64 * 128; i += 256) {
    int r = i >> 7, c = i & 127;
    long long rr = row0 + r;
    sA[r * 132 + c] = (rr < nrows) ? A[rr * 128 + c] : 0.f;
  }
  __syncthreads();

  const int wave = tid >> 5, lane = tid & 31;
  const int sel = lane >> 4;  // half-wave select
  const int l16 = lane & 15;
  const int cb = wave << 4;  // column base of this wave's tile
  const int col = cb + l16;

  for (int rt = 0; rt < 4; ++rt) {
    v8f acc = {};
    const int m = rt * 16 + l16;
#pragma unroll
    for (int k0 = 0; k0 < 128; k0 += 4) {
      const int ka = k0 + 2 * sel;
      v2f a, b;
      a[0] = sA[m * 132 + ka];
      a[1] = sA[m * 132 + ka + 1];
      b[0] = sW[ka * 128 + col];
      b[1] = sW[(ka + 1) * 128 + col];
      acc = wmma_k4(a, b, acc);
    }
#pragma unroll
    for (int i = 0; i < 8; ++i) {
      long long r = row0 + rt * 16 + i + 8 * sel;
      if (r < nrows) {
        float v = acc[i];
        if (EPI == EPI_GELU) {
          v += bias[col];
          v = 0.5f * v * (1.f + erff(v * 0.70710678118654752f));
        } else if (EPI == EPI_RES) {
          v += bias[col] + res[r * 128 + col];
        }
        out[r * 128 + col] = v;
      }
    }
  }
}

// ---------------------------------------------------------------------------
// a_src[n,h] = sum_c xh[n,h,c]*att_src[h,c];  a_dst likewise
// Grid is exact: (N_NODES*8)/256 blocks, no bounds check needed.
// ---------------------------------------------------------------------------
__global__ __launch_bounds__(256) void node_attn_kernel(
    const float* __restrict__ xh, const float* __restrict__ att_src,
    const float* __restrict__ att_dst, float* __restrict__ a_src,
    float* __restrict__ a_dst) {
  int t = blockIdx.x * 256 + threadIdx.x;
  int n = t >> 3, h = t & 7;
  const float4* xp = (const float4*)(xh + (long long)n * 128 + h * 16);
  const float4* sp = (const float4*)(att_src + h * 16);
  const float4* dp = (const float4*)(att_dst + h * 16);
  float as = 0.f, ad = 0.f;
#pragma unroll
  for (int j = 0; j < 4; ++j) {
    float4 xv = xp[j], sv = sp[j], dv = dp[j];
    as += xv.x * sv.x + xv.y * sv.y + xv.z * sv.z + xv.w * sv.w;
    ad += xv.x * dv.x + xv.y * dv.y + xv.z * dv.z + xv.w * dv.w;
  }
  a_src[t] = as;
  a_dst[t] = ad;
}

// ---------------------------------------------------------------------------
// w_att[k][h] = sum_c lin_edge_W[k, h*16+c] * att_edge[h, c]   (16 x 16,
// cols 8..15 zero-padded).  Folds the <e, att_edge> contraction into a
// precomputed K x H matrix so edge_alpha needs no per-head reduction.
// ---------------------------------------------------------------------------
__global__ __launch_bounds__(256) void attw_precompute_kernel(
    const float* __restrict__ lin_edge_W, const float* __restrict__ att_edge,
    float* __restrict__ w_att) {
  int t = threadIdx.x;
  if (t < 16 * 16) {
    int k = t >> 4, h = t & 15;
    float v = 0.f;
    if (h < 8) {
#pragma unroll
      for (int c = 0; c < 16; ++c)
        v += lin_edge_W[k * 128 + h * 16 + c] * att_edge[h * 16 + c];
    }
    w_att[t] = v;
  }
}

// ---------------------------------------------------------------------------
// edge_alpha: one WMMA chain gives alpha_e[16 edges][8 heads] directly
// (A = edge_attr tile 16x16, B = w_att 16x16).  C-layout: lane = head,
// VGPR i = edge row -> finalize straight from accumulator registers.
// Block = 256 threads (8 waves), wave w owns edge tile w -> 128 edges/block.
// Grid is exact (N_EDGES/128): no edge bounds checks.
// ---------------------------------------------------------------------------
__global__ __launch_bounds__(256) void edge_alpha_kernel(
    const float* __restrict__ edge_attr, const float* __restrict__ w_att,
    const float* __restrict__ a_src, const float* __restrict__ a_dst,
    const long long* __restrict__ srcI, const long long* __restrict__ dstI,
    float* __restrict__ alpha, unsigned int* __restrict__ m_enc) {
  __shared__ float sEA[128][20];  // pad 4 -> conflict-free A frags
  __shared__ float sB[16][16];
  const int tid = threadIdx.x;
  const long long eb = (long long)blockIdx.x * 128;

  if (tid < 16 * 16) sB[tid >> 4][tid & 15] = w_att[tid];
  for (int i = tid; i < 128 * 16; i += 256) {
    int r = i >> 4, c = i & 15;
    sEA[r][c] = edge_attr[(eb + r) * 16 + c];
  }
  __syncthreads();

  const int wave = tid >> 5, lane = tid & 31;
  const int sel = lane >> 4, l16 = lane & 15;
  const int r0 = wave * 16;

  v8f acc = {};
#pragma unroll
  for (int k0 = 0; k0 < 16; k0 += 4) {
    const int ka = k0 + 2 * sel;
    v2f a, b;
    a[0] = sEA[r0 + l16][ka];
    a[1] = sEA[r0 + l16][ka + 1];
    b[0] = sB[ka][l16];
    b[1] = sB[ka + 1][l16];
    acc = wmma_k4(a, b, acc);
  }

  if (l16 < 8) {  // valid heads only
    const int h = l16;
#pragma unroll
    for (int i = 0; i < 8; ++i) {
      long long e = eb + r0 + i + 8 * sel;
      long long s = srcI[e], d = dstI[e];
      float al = a_src[s * 8 + h] + a_dst[d * 8 + h] + acc[i];
      al = (al >= 0.f) ? al : NEG_SLOPE * al;
      alpha[e * 8 + h] = al;
      unsigned int u = __float_as_uint(al);
      unsigned int key = (u & 0x80000000u) ? ~u : (u | 0x80000000u);
      atomicMax(&m_enc[d * 8 + h], key);  // init 0 < any finite encoding
    }
  }
}

// ---------------------------------------------------------------------------
// ex = exp(alpha - m[dst]); s[dst] += ex  (alpha overwritten with ex)
// Grid is exact (N_EDGES*8/256): no bounds check.
// ---------------------------------------------------------------------------
__global__ __launch_bounds__(256) void edge_exp_kernel(
    float* __restrict__ alpha, const long long* __restrict__ dstI,
    const unsigned int* __restrict__ m_enc, float* __restrict__ ssum) {
  long long t = (long long)blockIdx.x * 256 + threadIdx.x;
  long long e = t >> 3;
  int h = (int)(t & 7);
  long long d = dstI[e];
  unsigned int key = m_enc[d * 8 + h];
  unsigned int u = (key & 0x80000000u) ? (key & 0x7FFFFFFFu) : ~key;
  float m = __uint_as_float(u);
  float ex = __expf(alpha[t] - m);
  alpha[t] = ex;
  atomicAdd(&ssum[d * 8 + h], ex);
}

// ---------------------------------------------------------------------------
// scatter: agg[dst, col] += (ex/(s[dst]+eps)) * xh[src,col] * e[edge,col]
// e recomputed from register-resident W column (16 FMAs). 16 edges/block.
// Grid is exact (N_EDGES/16): no bounds checks.
// ---------------------------------------------------------------------------
__global__ __launch_bounds__(256) void edge_scatter_kernel(
    const float* __restrict__ edge_attr, const float* __restrict__ lin_edge_W,
    const float* __restrict__ ex, const float* __restrict__ ssum,
    const long long* __restrict__ srcI, const long long* __restrict__ dstI,
    const float* __restrict__ xh, float* __restrict__ agg) {
  __shared__ float sEA[16][16];
  const int tid = threadIdx.x;
  const long long eb = (long long)blockIdx.x * 16;
  const int col = tid & 127;
  const int sub = tid >> 7;  // 0..1
  const int h = col >> 4;

  float wc[16];
#pragma unroll
  for (int k = 0; k < 16; ++k) wc[k] = lin_edge_W[k * 128 + col];

  {
    int r = tid >> 4, c = tid & 15;
    sEA[r][c] = edge_attr[(eb + r) * 16 + c];
  }
  __syncthreads();

  for (int it = 0; it < 8; ++it) {
    int r = sub + it * 2;
    long long e = eb + r;
    float ev = 0.f;
#pragma unroll
    for (int k = 0; k < 16; ++k) ev = fmaf(sEA[r][k], wc[k], ev);
    long long s = srcI[e], d = dstI[e];
    float a = ex[e * 8 + h] / (ssum[d * 8 + h] + 1e-16f);
    float msg = a * xh[s * 128 + col] * ev;
    atomicAdd(&agg[d * 128 + col], msg);
  }
}

// ---------------------------------------------------------------------------
// LayerNorm over 128 cols; one wave per row. v = a (+ acc) (+ bias)
// Grid is exact (N_NODES/8): no row guard.
// ---------------------------------------------------------------------------
__global__ __launch_bounds__(256) void ln_kernel(
    const float* __restrict__ a, const float* __restrict__ acc,
    const float* __restrict__ bias, const float* __restrict__ g,
    const float* __restrict__ b, float* __restrict__ out) {
  const int wave = threadIdx.x >> 5, lane = threadIdx.x & 31;
  const long long row = (long long)blockIdx.x * 8 + wave;
  const int c0 = lane * 4;
  float4 v = *(const float4*)(a + row * 128 + c0);
  if (acc) {
    float4 w = *(const float4*)(acc + row * 128 + c0);
    v.x += w.x; v.y += w.y; v.z += w.z; v.w += w.w;
  }
  if (bias) {
    float4 w = *(const float4*)(bias + c0);
    v.x += w.x; v.y += w.y; v.z += w.z; v.w += w.w;
  }
  float s = v.x + v.y + v.z + v.w;
#pragma unroll
  for (int m = 16; m >= 1; m >>= 1) s += __shfl_xor(s, m, 32);
  const float mu = s * (1.f / 128.f);
  float dx = v.x - mu, dy = v.y - mu, dz = v.z - mu, dw = v.w - mu;
  float q = dx * dx + dy * dy + dz * dz + dw * dw;
#pragma unroll
  for (int m = 16; m >= 1; m >>= 1) q += __shfl_xor(q, m, 32);
  const float inv = rsqrtf(q * (1.f / 128.f) + LN_EPS);
  float4 gv = *(const float4*)(g + c0);
  float4 bv = *(const float4*)(b + c0);
  float4 o;
  o.x = dx * inv * gv.x + bv.x;
  o.y = dy * inv * gv.y + bv.y;
  o.z = dz * inv * gv.z + bv.z;
  o.w = dw * inv * gv.w + bv.w;
  *(float4*)(out + row * 128 + c0) = o;
}

// ---------------------------------------------------------------------------
extern "C" void kernel_launch(void* const* d_in, const int* in_sizes, int n_in,
                              void* d_out, int out_size, void* d_ws,
                              size_t ws_size, hipStream_t stream) {
  const float* x = (const float*)d_in[0];
  const long long* edge_index = (const long long*)d_in[1];  // int64 in reference
  const float* edge_attr = (const float*)d_in[2];
  const float* lin_W = (const float*)d_in[3];
  const float* att_src = (const float*)d_in[4];
  const float* att_dst = (const float*)d_in[5];
  const float* lin_edge_W = (const float*)d_in[6];
  const float* att_edge = (const float*)d_in[7];
  const float* conv_bias = (const float*)d_in[8];
  const float* ln1_g = (const float*)d_in[9];
  const float* ln1_b = (const float*)d_in[10];
  const float* W1 = (const float*)d_in[11];
  const float* b1 = (const float*)d_in[12];
  const float* W2 = (const float*)d_in[13];
  const float* b2 = (const float*)d_in[14];
  const float* ln2_g = (const float*)d_in[15];
  const float* ln2_b = (const float*)d_in[16];
  float* out = (float*)d_out;

  const long long* srcI = edge_index;
  const long long* dstI = edge_index + N_EDGES;

  // workspace carving (~218 MB total)
  char* w = (char*)d_ws;
  auto carve = [&](size_t bytes) {
    char* p = w;
    w += (bytes + 255) & ~(size_t)255;
    return p;
  };
  float* xh = (float*)carve((size_t)N_NODES * 128 * 4);
  float* a_src = (float*)carve((size_t)N_NODES * 8 * 4);
  float* a_dst = (float*)carve((size_t)N_NODES * 8 * 4);
  float* alpha = (float*)carve((size_t)N_EDGES * 8 * 4);  // alpha -> ex
  unsigned int* m_enc = (unsigned int*)carve((size_t)N_NODES * 8 * 4);
  float* ssum = (float*)carve((size_t)N_NODES * 8 * 4);
  float* agg = (float*)carve((size_t)N_NODES * 128 * 4);
  float* hbuf = (float*)carve((size_t)N_NODES * 128 * 4);
  float* w_att = (float*)carve((size_t)16 * 16 * 4);
  float* z1 = agg;   // reuse: agg dead after LN1
  float* ybuf = xh;  // reuse: xh dead after scatter

  (void)hipMemsetAsync(m_enc, 0, (size_t)N_NODES * 8 * 4, stream);
  (void)hipMemsetAsync(ssum, 0, (size_t)N_NODES * 8 * 4, stream);
  (void)hipMemsetAsync(agg, 0, (size_t)N_NODES * 128 * 4, stream);

  const size_t gemm_lds = (size_t)(128 * 128 + 64 * 132) * sizeof(float);
  const int gemm_blocks = (N_NODES + 63) / 64;

  // 0) fold att_edge into lin_edge_W -> w_att[16][16]
  attw_precompute_kernel<<<1, 256, 0, stream>>>(lin_edge_W, att_edge, w_att);
  // 1) xh = x @ lin_W
  gemm128_kernel<EPI_NONE><<<gemm_blocks, 256, gemm_lds, stream>>>(
      x, lin_W, nullptr, nullptr, xh, N_NODES);
  // 2) per-(node,head) attention coefficients
  node_attn_kernel<<<N_NODES * 8 / 256, 256, 0, stream>>>(xh, att_src, att_dst,
                                                          a_src, a_dst);
  // 3) raw alpha + segment max (single WMMA chain per 16-edge tile)
  edge_alpha_kernel<<<N_EDGES / 128, 256, 0, stream>>>(
      edge_attr, w_att, a_src, a_dst, srcI, dstI, alpha, m_enc);
  // 4) exp + segment sum
  edge_exp_kernel<<<(int)((long long)N_EDGES * 8 / 256), 256, 0, stream>>>(
      alpha, dstI, m_enc, ssum);
  // 5) normalized message scatter-add
  edge_scatter_kernel<<<N_EDGES / 16, 256, 0, stream>>>(
      edge_attr, lin_edge_W, alpha, ssum, srcI, dstI, xh, agg);
  // 6) h = LN1(x + agg + conv_bias)
  ln_kernel<<<N_NODES / 8, 256, 0, stream>>>(x, agg, conv_bias, ln1_g, ln1_b,
                                             hbuf);
  // 7) z1 = gelu(h @ W1 + b1)
  gemm128_kernel<EPI_GELU><<<gemm_blocks, 256, gemm_lds, stream>>>(
      hbuf, W1, b1, nullptr, z1, N_NODES);
  // 8) y = h + z1 @ W2 + b2
  gemm128_kernel<EPI_RES><<<gemm_blocks, 256, gemm_lds, stream>>>(
      z1, W2, b2, hbuf, ybuf, N_NODES);
  // 9) out = LN2(y)
  ln_kernel<<<N_NODES / 8, 256, 0, stream>>>(ybuf, nullptr, nullptr, ln2_g,
                                             ln2_b, out);
}